// SigmaWrapper_3169685864818
// MI455X (gfx1250) — compile-verified
//
#include <hip/hip_runtime.h>

typedef float v2f __attribute__((ext_vector_type(2)));
typedef float v8f __attribute__((ext_vector_type(8)));
typedef unsigned int u32x4 __attribute__((ext_vector_type(4)));
typedef int i32x4 __attribute__((ext_vector_type(4)));
typedef int i32x8 __attribute__((ext_vector_type(8)));

#define B_ROWS 8192
#define D_IN   4096
#define D_OUT  4096
#define EPS_F  1e-4f

#define TM 128
#define TN 128
#define TK 16

#define LIN_ST 18   // linear kernel: M-major LDS rows of 16 K-floats + 2 pad (even -> b64-aligned)
#define XTX_ST 132  // hess kernel: K-major LDS rows of 128 M-floats + 4 pad (528B, 16B-aligned)

// ---------------------------------------------------------------------------
// Tensor Data Mover: issue a 2-D tile load (global -> LDS) via the TDM D#.
// D# group0: [1:0]=count=1, [63:32]=lds_addr, [120:64]=global_addr, [127:126]=type=2
// D# group1: [17:16]=data_size(2 -> 4B), [20]=pad_enable, [24:22]=pad_interval,
//            [31:25]=pad_amount, [79:48]=tensor_dim0, [111:80]=tensor_dim1,
//            [127:112]=tile_dim0, [143:128]=tile_dim1, [207:160]=tensor_dim0_stride
// groups 2/3 (and the extra group of the 6-arg builtin) zero for a 2-D tile.
// Tracked by TENSORcnt. pad_interval code p pads every 2^(p+1) DWORDs;
// pad_amount code a pads a+1 DWORDs.
// ---------------------------------------------------------------------------
__device__ __forceinline__ void tdm_load_2d(
    unsigned lds_byte_off, const float* gptr,
    unsigned tile_d0, unsigned tile_d1,
    unsigned tensor_d0, unsigned tensor_d1, unsigned stride_d0_elems,
    unsigned pad_interval_code, unsigned pad_amount_code)
{
  const unsigned long long ga = (unsigned long long)(uintptr_t)gptr;
  u32x4 g0;
  g0.x = 1u;                                                   // count=1, user descriptor
  g0.y = lds_byte_off;                                         // LDS byte address
  g0.z = (unsigned)ga;                                         // global_addr[31:0]
  g0.w = (unsigned)((ga >> 32) & 0x01ffffffu) | (2u << 30);    // global_addr[56:32] | type=2

  i32x8 g1;
  g1[0] = (int)((2u << 16) | (1u << 20) |
                (pad_interval_code << 22) | (pad_amount_code << 25));
  g1[1] = (int)((tensor_d0 & 0xffffu) << 16);                  // tensor_dim0[15:0]
  g1[2] = (int)((tensor_d0 >> 16) | ((tensor_d1 & 0xffffu) << 16));
  g1[3] = (int)((tensor_d1 >> 16) | (tile_d0 << 16));          // tile_dim0
  g1[4] = (int)(tile_d1 & 0xffffu);                            // tile_dim1 (tile_dim2=0)
  g1[5] = (int)stride_d0_elems;                                // tensor_dim0_stride[31:0]
  g1[6] = 0;                                                   // stride hi + dim1_stride lo
  g1[7] = 0;

  i32x4 z4 = {0, 0, 0, 0};
  i32x8 z8 = {0, 0, 0, 0, 0, 0, 0, 0};
  __builtin_amdgcn_tensor_load_to_lds(g0, g1, z4, z4, z8, 0);
}

// ---------------------------------------------------------------------------
// WMMA compute phases: 8-wave workgroup; wave (wm,wn) owns a 32x64 subtile as
// 2x4 16x16 f32 accumulators. Fragment layout per CDNA5 16x16x4 F32 spec:
// lane l -> m/n = l%16, half-wave (l/16) selects K pair {0,1} vs {2,3}.
// ---------------------------------------------------------------------------
__device__ __forceinline__ void wmma_accum_kmajor(
    const float* __restrict__ As, const float* __restrict__ Bs,
    int m0, int n0, int ln, int kg, v8f acc[2][4])
{
#pragma unroll
  for (int kk = 0; kk < TK; kk += 4) {
    v2f a[2], b[4];
#pragma unroll
    for (int i = 0; i < 2; ++i) {
      a[i].x = As[(kk + 2 * kg + 0) * XTX_ST + m0 + i * 16 + ln];
      a[i].y = As[(kk + 2 * kg + 1) * XTX_ST + m0 + i * 16 + ln];
    }
#pragma unroll
    for (int j = 0; j < 4; ++j) {
      b[j].x = Bs[(kk + 2 * kg + 0) * XTX_ST + n0 + j * 16 + ln];
      b[j].y = Bs[(kk + 2 * kg + 1) * XTX_ST + n0 + j * 16 + ln];
    }
#pragma unroll
    for (int i = 0; i < 2; ++i)
#pragma unroll
      for (int j = 0; j < 4; ++j)
        acc[i][j] = __builtin_amdgcn_wmma_f32_16x16x4_f32(
            false, a[i], false, b[j], (short)0, acc[i][j], false, false);
  }
}

__device__ __forceinline__ void wmma_accum_mmajor(
    const float* __restrict__ As, const float* __restrict__ Bs,
    int m0, int n0, int ln, int kg, v8f acc[2][4])
{
#pragma unroll
  for (int kk = 0; kk < TK; kk += 4) {
    v2f a[2], b[4];
#pragma unroll
    for (int i = 0; i < 2; ++i)
      a[i] = *(const v2f*)&As[(m0 + i * 16 + ln) * LIN_ST + kk + 2 * kg];
#pragma unroll
    for (int j = 0; j < 4; ++j)
      b[j] = *(const v2f*)&Bs[(n0 + j * 16 + ln) * LIN_ST + kk + 2 * kg];
#pragma unroll
    for (int i = 0; i < 2; ++i)
#pragma unroll
      for (int j = 0; j < 4; ++j)
        acc[i][j] = __builtin_amdgcn_wmma_f32_16x16x4_f32(
            false, a[i], false, b[j], (short)0, acc[i][j], false, false);
  }
}

// ---------------------------------------------------------------------------
// out[m,n] = sum_k X[m,k]*W[n,k] + bias[n]
// TDM tiles: 16 contiguous K-floats x 128 rows, padded to stride-18 M-major LDS.
// ---------------------------------------------------------------------------
__global__ __launch_bounds__(256) void linear_wmma_kernel(
    const float* __restrict__ X, const float* __restrict__ W,
    const float* __restrict__ bias, float* __restrict__ out)
{
  __shared__ float As[2][TM * LIN_ST];
  __shared__ float Bs[2][TN * LIN_ST];

  const int tid  = threadIdx.x;
  const int mblk = blockIdx.y * TM;
  const int nblk = blockIdx.x * TN;

  const int w  = tid >> 5;
  const int l  = tid & 31;
  const int ln = l & 15;
  const int kg = l >> 4;
  const int m0 = (w & 3) * 32;
  const int n0 = (w >> 2) * 64;

  v8f acc[2][4] = {};

  const float* Abase = X + (size_t)mblk * D_IN;
  const float* Bbase = W + (size_t)nblk * D_IN;

  if (tid < 32) {  // one wave drives the TDM
    tdm_load_2d((unsigned)(uintptr_t)&As[0][0], Abase, TK, TM, D_IN, B_ROWS, D_IN, 3, 1);
    tdm_load_2d((unsigned)(uintptr_t)&Bs[0][0], Bbase, TK, TN, D_IN, D_OUT, D_IN, 3, 1);
  }

  for (int kblk = 0; kblk < D_IN; kblk += TK) {
    const int buf = (kblk / TK) & 1;
    if (tid < 32) {
      if (kblk + TK < D_IN) {
        tdm_load_2d((unsigned)(uintptr_t)&As[buf ^ 1][0], Abase + kblk + TK,
                    TK, TM, D_IN, B_ROWS, D_IN, 3, 1);
        tdm_load_2d((unsigned)(uintptr_t)&Bs[buf ^ 1][0], Bbase + kblk + TK,
                    TK, TN, D_IN, D_OUT, D_IN, 3, 1);
        __builtin_amdgcn_s_wait_tensorcnt(2);  // current buffer's 2 loads complete
      } else {
        __builtin_amdgcn_s_wait_tensorcnt(0);
      }
    }
    __syncthreads();
    wmma_accum_mmajor(&As[buf][0], &Bs[buf][0], m0, n0, ln, kg, acc);
    __syncthreads();
  }

#pragma unroll
  for (int i = 0; i < 2; ++i) {
#pragma unroll
    for (int j = 0; j < 4; ++j) {
      const int col = nblk + n0 + j * 16 + ln;
      const float bv = bias[col];
#pragma unroll
      for (int v = 0; v < 8; ++v) {
        const int row = mblk + m0 + i * 16 + kg * 8 + v;  // C: M = vgpr + 8*(l/16)
        out[(size_t)row * D_OUT + col] = acc[i][j][v] + bv;
      }
    }
  }
}

// ---------------------------------------------------------------------------
// hess[i,j] = sum_b X[b,i]*X[b,j] + EPS*(i==j)
// TDM tiles: 128 contiguous M/N-floats x 16 K-rows -> stride-132 K-major LDS.
// ---------------------------------------------------------------------------
__global__ __launch_bounds__(256) void xtx_wmma_kernel(
    const float* __restrict__ X, float* __restrict__ hess)
{
  __shared__ float As[2][TK * XTX_ST];
  __shared__ float Bs[2][TK * XTX_ST];

  const int tid  = threadIdx.x;
  const int mblk = blockIdx.y * TM;
  const int nblk = blockIdx.x * TN;

  const int w  = tid >> 5;
  const int l  = tid & 31;
  const int ln = l & 15;
  const int kg = l >> 4;
  const int m0 = (w & 3) * 32;
  const int n0 = (w >> 2) * 64;

  v8f acc[2][4] = {};

  if (tid < 32) {
    tdm_load_2d((unsigned)(uintptr_t)&As[0][0], X + mblk, TM, TK, D_IN, B_ROWS, D_IN, 6, 3);
    tdm_load_2d((unsigned)(uintptr_t)&Bs[0][0], X + nblk, TM, TK, D_IN, B_ROWS, D_IN, 6, 3);
  }

  for (int kblk = 0; kblk < B_ROWS; kblk += TK) {
    const int buf = (kblk / TK) & 1;
    if (tid < 32) {
      if (kblk + TK < B_ROWS) {
        tdm_load_2d((unsigned)(uintptr_t)&As[buf ^ 1][0],
                    X + (size_t)(kblk + TK) * D_IN + mblk,
                    TM, TK, D_IN, B_ROWS, D_IN, 6, 3);
        tdm_load_2d((unsigned)(uintptr_t)&Bs[buf ^ 1][0],
                    X + (size_t)(kblk + TK) * D_IN + nblk,
                    TM, TK, D_IN, B_ROWS, D_IN, 6, 3);
        __builtin_amdgcn_s_wait_tensorcnt(2);
      } else {
        __builtin_amdgcn_s_wait_tensorcnt(0);
      }
    }
    __syncthreads();
    wmma_accum_kmajor(&As[buf][0], &Bs[buf][0], m0, n0, ln, kg, acc);
    __syncthreads();
  }

#pragma unroll
  for (int i = 0; i < 2; ++i) {
#pragma unroll
    for (int j = 0; j < 4; ++j) {
      const int col = nblk + n0 + j * 16 + ln;
#pragma unroll
      for (int v = 0; v < 8; ++v) {
        const int row = mblk + m0 + i * 16 + kg * 8 + v;
        float val = acc[i][j][v];
        if (row == col) val += EPS_F;
        hess[(size_t)row * D_IN + col] = val;
      }
    }
  }
}

extern "C" void kernel_launch(void* const* d_in, const int* in_sizes, int n_in,
                              void* d_out, int out_size, void* d_ws, size_t ws_size,
                              hipStream_t stream) {
  (void)in_sizes; (void)n_in; (void)out_size; (void)d_ws; (void)ws_size;
  const float* x = (const float*)d_in[0];   // [B, D]
  const float* W = (const float*)d_in[1];   // [DOUT, D]
  const float* b = (const float*)d_in[2];   // [DOUT]
  float* out  = (float*)d_out;                           // [B, DOUT]
  float* hess = (float*)d_out + (size_t)B_ROWS * D_OUT;  // [D, D]

  dim3 blk(256);
  dim3 g1(D_OUT / TN, B_ROWS / TM);
  linear_wmma_kernel<<<g1, blk, 0, stream>>>(x, W, b, out);

  dim3 g2(D_IN / TN, D_IN / TM);
  xtx_wmma_kernel<<<g2, blk, 0, stream>>>(x, hess);
}